// MemoryCell_876173328393
// MI455X (gfx1250) — compile-verified
//
#include <hip/hip_runtime.h>
#include <hip/hip_bf16.h>

// LTC / NCP MemoryCell scan for MI455X (gfx1250, wave32).
//
// Key structural insight: src_idx[u] = {u, (u+dim) % U} means unit u and unit
// u+dim only ever read each other's state, and both read the same sensory
// input x[b, u % dim].  The entire T=128 scan therefore decomposes into
// B*dim = 32768 fully independent 2-state ODE systems.  One thread owns one
// pair: both states and all (softplus-prefolded) parameters stay in VGPRs for
// the whole scan.  No LDS, no barriers, no inter-thread traffic.
//
// The workload is transcendental-VALU bound (~100M v_exp_f32, ~150M v_rcp_f32)
// and reads only ~17 MB once (<1 us at 23.3 TB/s HBM) -> WMMA is inapplicable
// (no matrix product exists in the op); the right CDNA5 levers are wave32
// occupancy (1024 waves), register-resident recurrent state, fast TRANS ops,
// and global_prefetch for the streamed input.

#define DIMX   1024
#define BATCH  32
#define TSTEPS 128
#define NUNF   6
#define EPSV   1e-8f

__device__ __forceinline__ float fast_rcp(float x) {
    return __builtin_amdgcn_rcpf(x);            // v_rcp_f32, ~1 ulp
}
__device__ __forceinline__ float fast_sigmoid(float z) {
    return fast_rcp(1.0f + __expf(-z));         // v_exp_f32 + v_rcp_f32
}
__device__ __forceinline__ float softplusf(float x) {
    // inputs to softplus here are in (0,1); plain form is exact enough
    return log1pf(__expf(x));
}

__global__ __launch_bounds__(256) void ltc_pair_scan_kernel(
    const float* __restrict__ inp,        // (B, T, dim)
    const float* __restrict__ gleak,      // (U,)
    const float* __restrict__ vleak,      // (U,)
    const float* __restrict__ cm,         // (U,)
    const float* __restrict__ w,          // (U, 2)
    const float* __restrict__ sigma,      // (U, 2)
    const float* __restrict__ mu,         // (U, 2)
    const float* __restrict__ erev,       // (U, 2)
    const float* __restrict__ sens_w,     // (U,)
    const float* __restrict__ sens_sigma, // (U,)
    const float* __restrict__ sens_mu,    // (U,)
    const float* __restrict__ sens_erev,  // (U,)
    const float* __restrict__ input_w,    // (dim,)
    const float* __restrict__ input_b,    // (dim,)
    const float* __restrict__ output_w,   // (dim,)
    const float* __restrict__ output_b,   // (dim,)
    float* __restrict__ out)              // (B, dim)
{
    const int tid = blockIdx.x * blockDim.x + threadIdx.x;   // 0 .. B*dim-1
    const int b  = tid >> 10;          // / DIMX
    const int u  = tid & (DIMX - 1);   // % DIMX
    const int ua = u;                  // unit a  (u < dim)
    const int ub = u + DIMX;           // unit b  (partner, u+dim)

    // ---- fold parameters into registers (softplus applied once) ----------
    const float gl_a  = softplusf(gleak[ua]);
    const float gl_b  = softplusf(gleak[ub]);
    const float vl_a  = vleak[ua];
    const float vl_b  = vleak[ub];
    const float cmt_a = softplusf(cm[ua]) * (float)NUNF;
    const float cmt_b = softplusf(cm[ub]) * (float)NUNF;

    const float spw_a0 = softplusf(w[2 * ua + 0]);
    const float spw_a1 = softplusf(w[2 * ua + 1]);
    const float spw_b0 = softplusf(w[2 * ub + 0]);
    const float spw_b1 = softplusf(w[2 * ub + 1]);

    const float sig_a0 = sigma[2 * ua + 0], sig_a1 = sigma[2 * ua + 1];
    const float sig_b0 = sigma[2 * ub + 0], sig_b1 = sigma[2 * ub + 1];
    const float mu_a0  = mu[2 * ua + 0],    mu_a1  = mu[2 * ua + 1];
    const float mu_b0  = mu[2 * ub + 0],    mu_b1  = mu[2 * ub + 1];
    const float er_a0  = erev[2 * ua + 0],  er_a1  = erev[2 * ua + 1];
    const float er_b0  = erev[2 * ub + 0],  er_b1  = erev[2 * ub + 1];

    const float ssw_a  = softplusf(sens_w[ua]);
    const float ssw_b  = softplusf(sens_w[ub]);
    const float ssig_a = sens_sigma[ua], ssig_b = sens_sigma[ub];
    const float smu_a  = sens_mu[ua],    smu_b  = sens_mu[ub];
    const float serev_a = sens_erev[ua], serev_b = sens_erev[ub];

    const float in_w  = input_w[u];
    const float in_b  = input_b[u];
    const float ow    = output_w[u];
    const float ob    = output_b[u];

    // hoisted per-thread constants
    const float glvl_a     = gl_a * vl_a;
    const float glvl_b     = gl_b * vl_b;
    const float denbase_a  = cmt_a + gl_a + EPSV;
    const float denbase_b  = cmt_b + gl_b + EPSV;

    const float* __restrict__ xin =
        inp + (size_t)b * (size_t)TSTEPS * (size_t)DIMX + (size_t)u;

    float va = 0.0f;   // v[u]
    float vb = 0.0f;   // v[u+dim]

    for (int t = 0; t < TSTEPS; ++t) {
        // stream-ahead prefetch of the input time series (global_prefetch_b8)
        if (t + 8 < TSTEPS)
            __builtin_prefetch(xin + (size_t)(t + 8) * DIMX, 0, 0);

        const float x = fmaf(xin[(size_t)t * DIMX], in_w, in_b);

        // sensory synapse: constant across the 6 unfolds -> hoist
        const float sact_a = ssw_a * fast_sigmoid((x - smu_a) * ssig_a);
        const float sact_b = ssw_b * fast_sigmoid((x - smu_b) * ssig_b);
        const float numbase_a = fmaf(sact_a, serev_a, glvl_a);
        const float numbase_b = fmaf(sact_b, serev_b, glvl_b);
        const float denT_a = denbase_a + sact_a;
        const float denT_b = denbase_b + sact_b;

#pragma unroll
        for (int k = 0; k < NUNF; ++k) {
            // unit a synapses read (va, vb); unit b synapses read (vb, va)
            const float wa0 = spw_a0 * fast_sigmoid((va - mu_a0) * sig_a0);
            const float wa1 = spw_a1 * fast_sigmoid((vb - mu_a1) * sig_a1);
            const float wb0 = spw_b0 * fast_sigmoid((vb - mu_b0) * sig_b0);
            const float wb1 = spw_b1 * fast_sigmoid((va - mu_b1) * sig_b1);

            const float num_a = fmaf(wa0, er_a0, fmaf(wa1, er_a1, numbase_a));
            const float num_b = fmaf(wb0, er_b0, fmaf(wb1, er_b1, numbase_b));
            const float den_a = denT_a + wa0 + wa1;
            const float den_b = denT_b + wb0 + wb1;

            const float na = fmaf(cmt_a, va, num_a);
            const float nb = fmaf(cmt_b, vb, num_b);
            va = na * fast_rcp(den_a);
            vb = nb * fast_rcp(den_b);
        }
    }

    out[(size_t)b * DIMX + u] = fmaf(va, ow, ob);
}

extern "C" void kernel_launch(void* const* d_in, const int* in_sizes, int n_in,
                              void* d_out, int out_size, void* d_ws, size_t ws_size,
                              hipStream_t stream) {
    (void)in_sizes; (void)n_in; (void)d_ws; (void)ws_size; (void)out_size;

    const float* inp        = (const float*)d_in[0];
    const float* gleak      = (const float*)d_in[1];
    const float* vleak      = (const float*)d_in[2];
    const float* cm         = (const float*)d_in[3];
    const float* w          = (const float*)d_in[4];
    const float* sigma      = (const float*)d_in[5];
    const float* mu         = (const float*)d_in[6];
    const float* erev       = (const float*)d_in[7];
    const float* sens_w     = (const float*)d_in[8];
    const float* sens_sigma = (const float*)d_in[9];
    const float* sens_mu    = (const float*)d_in[10];
    const float* sens_erev  = (const float*)d_in[11];
    const float* input_w    = (const float*)d_in[12];
    const float* input_b    = (const float*)d_in[13];
    const float* output_w   = (const float*)d_in[14];
    const float* output_b   = (const float*)d_in[15];
    float* out              = (float*)d_out;

    const int total  = BATCH * DIMX;        // 32768 threads = 1024 wave32s
    const int block  = 256;                 // 8 waves / workgroup
    const int grid   = total / block;       // 128 workgroups

    ltc_pair_scan_kernel<<<grid, block, 0, stream>>>(
        inp, gleak, vleak, cm, w, sigma, mu, erev,
        sens_w, sens_sigma, sens_mu, sens_erev,
        input_w, input_b, output_w, output_b, out);
}